// AttentionAggregator_62981400429066
// MI455X (gfx1250) — compile-verified
//
#include <hip/hip_runtime.h>
#include <hip/hip_bf16.h>
#include <math.h>

typedef float v2f __attribute__((ext_vector_type(2)));
typedef float v8f __attribute__((ext_vector_type(8)));

#define IN_DIM 256
#define OUT_DIM 64
#define SLOPE 0.1f
#define KC 128          // K-chunk staged in LDS
#define LDP 132         // padded LDS row stride in floats (132%64==4 -> no bank conflicts)

// ---------------------------------------------------------------------------
// Kernel 1: emb = features @ W^T + b   via V_WMMA_F32_16X16X4_F32
// block = 256 threads = 8 waves; block tile = 32 rows x 64 cols.
// wave w: row strip (w/4)*16, col tile (w%4)*16. K looped in 2 chunks of 128.
// ---------------------------------------------------------------------------
__global__ __launch_bounds__(256) void gat_gemm_wmma(
    const float* __restrict__ features,
    const float* __restrict__ W,
    const float* __restrict__ bias,
    float* __restrict__ emb,
    int n)
{
    __shared__ float ldsA[32 * LDP];       // features tile 32 x 128 (padded)
    __shared__ float ldsB[OUT_DIM * LDP];  // W tile       64 x 128 (padded)

    const int tid  = threadIdx.x;
    const int row0 = blockIdx.x * 32;

    const int wave  = tid >> 5;        // 0..7
    const int lane  = tid & 31;
    const int mbase = (wave >> 2) * 16;   // 0 or 16
    const int n0    = (wave & 3) * 16;    // 0,16,32,48
    const int lm    = lane & 15;          // row/col within fragment
    const int half  = lane >> 4;          // lane half
    const int kh    = half * 2;           // K sub-offset within a 4-wide step

    v8f acc = {0.f, 0.f, 0.f, 0.f, 0.f, 0.f, 0.f, 0.f};

    for (int kc = 0; kc < IN_DIM; kc += KC) {
        // stage A: 32 rows x 128 cols = 1024 float4, 4 per thread (coalesced)
        #pragma unroll
        for (int i = 0; i < 4; ++i) {
            int idx = tid + i * 256;          // 0..1023
            int r   = idx >> 5;               // 32 float4 per row
            int c4  = idx & 31;
            float4 v = make_float4(0.f, 0.f, 0.f, 0.f);
            int gr = row0 + r;
            if (gr < n)
                v = *(const float4*)(features + (size_t)gr * IN_DIM + kc + c4 * 4);
            *(float4*)(&ldsA[r * LDP + c4 * 4]) = v;
        }
        // stage B: 64 rows x 128 cols = 2048 float4, 8 per thread
        #pragma unroll
        for (int i = 0; i < 8; ++i) {
            int idx = tid + i * 256;          // 0..2047
            int r   = idx >> 5;
            int c4  = idx & 31;
            float4 v = *(const float4*)(W + (size_t)r * IN_DIM + kc + c4 * 4);
            *(float4*)(&ldsB[r * LDP + c4 * 4]) = v;
        }
        __syncthreads();

        // A frag: lane -> feat[mbase+lm][k+kh..k+kh+1]
        // B frag: lane -> W[n0+lm][k+kh..k+kh+1]   (B[k][n] = W[n][k])
        const float* pA = &ldsA[(mbase + lm) * LDP + kh];
        const float* pB = &ldsB[(n0   + lm) * LDP + kh];
        #pragma unroll 8
        for (int k = 0; k < KC; k += 4) {
            v2f afrag = *(const v2f*)(pA + k);
            v2f bfrag = *(const v2f*)(pB + k);
            acc = __builtin_amdgcn_wmma_f32_16x16x4_f32(
                false, afrag, false, bfrag, (short)0, acc, false, false);
        }
        __syncthreads();
    }

    // epilogue: C/D layout -> VGPR r: lanes 0-15 row r, lanes 16-31 row r+8
    const float bv = bias[n0 + lm];
    #pragma unroll
    for (int r = 0; r < 8; ++r) {
        int grow = row0 + mbase + r + half * 8;
        if (grow < n)
            emb[(size_t)grow * OUT_DIM + n0 + lm] = acc[r] + bv;
    }
}

// ---------------------------------------------------------------------------
// Kernel 2: per-node scalars s1 = emb.a1, s2 = emb.a2; fold the self-loop
// (src==dst==i) into out/row_sum initialization (no atomics needed for it).
// ---------------------------------------------------------------------------
__global__ __launch_bounds__(256) void gat_node(
    const float* __restrict__ emb,
    const float* __restrict__ a,
    float* __restrict__ s1,
    float* __restrict__ s2,
    float* __restrict__ rowsum,
    float* __restrict__ out,
    int n)
{
    int i = blockIdx.x * blockDim.x + threadIdx.x;
    if (i >= n) return;
    const float* er = emb + (size_t)i * OUT_DIM;
    float d1 = 0.f, d2 = 0.f;
    #pragma unroll
    for (int j = 0; j < OUT_DIM; ++j) {
        float e = er[j];
        d1 += e * a[j];
        d2 += e * a[OUT_DIM + j];
    }
    s1[i] = d1;
    s2[i] = d2;
    float z = d1 + d2;
    float l = z > 0.f ? z : SLOPE * z;
    float w = expf(l);
    rowsum[i] = w;                       // self-loop contribution
    float* orow = out + (size_t)i * OUT_DIM;
    #pragma unroll
    for (int j = 0; j < OUT_DIM; ++j)
        orow[j] = w * er[j];             // initializes d_out fully each call
}

// ---------------------------------------------------------------------------
// Kernel 3: one wave32 per edge. Each lane covers 2 consecutive columns:
// float2 load of emb[dst], two global f32 atomic adds into out[src].
// ---------------------------------------------------------------------------
__global__ __launch_bounds__(256) void gat_edge(
    const int* __restrict__ edges,
    const float* __restrict__ emb,
    const float* __restrict__ s1,
    const float* __restrict__ s2,
    float* __restrict__ rowsum,
    float* __restrict__ out,
    int E)
{
    int wid  = (blockIdx.x * blockDim.x + threadIdx.x) >> 5;
    int lane = threadIdx.x & 31;
    if (wid >= E) return;
    int src = edges[2 * wid];
    int dst = edges[2 * wid + 1];
    float z = s1[src] + s2[dst];
    float l = z > 0.f ? z : SLOPE * z;
    float w = expf(l);
    if (lane == 0) atomicAdd(rowsum + src, w);
    const float2 ev = *(const float2*)(emb + (size_t)dst * OUT_DIM + 2 * lane);
    float* op = out + (size_t)src * OUT_DIM + 2 * lane;
    atomicAdd(op,     w * ev.x);
    atomicAdd(op + 1, w * ev.y);
}

// ---------------------------------------------------------------------------
// Kernel 4: out /= row_sum (broadcast per row of 64)
// ---------------------------------------------------------------------------
__global__ __launch_bounds__(256) void gat_final(
    float* __restrict__ out,
    const float* __restrict__ rowsum,
    int total)
{
    int idx = blockIdx.x * blockDim.x + threadIdx.x;
    if (idx >= total) return;
    out[idx] /= rowsum[idx >> 6];        // OUT_DIM == 64
}

extern "C" void kernel_launch(void* const* d_in, const int* in_sizes, int n_in,
                              void* d_out, int out_size, void* d_ws, size_t ws_size,
                              hipStream_t stream)
{
    (void)n_in; (void)out_size; (void)ws_size;
    const float* features = (const float*)d_in[0];
    const float* W        = (const float*)d_in[1];
    const float* bias     = (const float*)d_in[2];
    const float* a        = (const float*)d_in[3];
    const int*   edges    = (const int*)d_in[4];

    const int n = in_sizes[0] / IN_DIM;   // 100000
    const int E = in_sizes[4] / 2;        // 1600000
    float* out = (float*)d_out;

    // workspace: emb (n*64) | s1 (n) | s2 (n) | rowsum (n)  ~ 26.8 MB
    float* emb    = (float*)d_ws;
    float* s1     = emb + (size_t)n * OUT_DIM;
    float* s2     = s1 + n;
    float* rowsum = s2 + n;

    gat_gemm_wmma<<<(n + 31) / 32, 256, 0, stream>>>(features, W, bias, emb, n);
    gat_node<<<(n + 255) / 256, 256, 0, stream>>>(emb, a, s1, s2, rowsum, out, n);
    gat_edge<<<(E + 7) / 8, 256, 0, stream>>>(edges, emb, s1, s2, rowsum, out, E);
    const int total = n * OUT_DIM;
    gat_final<<<(total + 255) / 256, 256, 0, stream>>>(out, rowsum, total);
}